// KNNGraph_37864431682013
// MI455X (gfx1250) — compile-verified
//
#include <hip/hip_runtime.h>

// KNN graph feature builder for MI455X (gfx1250, wave32).
// Phase 1: WMMA f32 16x16x4 distance tiles -> per-thread register top-21 -> idx table in d_ws
// Phase 2: coalesced gather/concat writes (store-BW bound, ~335MB @ 23.3TB/s)

#define B_    8
#define C_    64
#define N_    4096
#define KNN_  21
#define KOUT  20
#define QPB   128            // queries per block (phase 1)
#define WAVES 4              // 128 threads = 4 wave32
#define QPW   32             // queries per wave (two 16-row WMMA tiles)

typedef __attribute__((ext_vector_type(2))) float v2f;
typedef __attribute__((ext_vector_type(8))) float v8f;

__global__ __launch_bounds__(QPB) void knn_topk_kernel(
    const float* __restrict__ points, int* __restrict__ knn_idx)
{
    __shared__ float s_p[3 * N_];            // x[0..N) | y[N..2N) | z[2N..3N)
    __shared__ float s_sq[N_];
    __shared__ float s_dt[WAVES][QPW][17];   // padded to dodge bank conflicts

    const int blocks_per_batch = N_ / QPB;                // 32
    const int b      = blockIdx.x / blocks_per_batch;
    const int qchunk = blockIdx.x % blocks_per_batch;
    const int tid    = threadIdx.x;

    const float* px = points + (size_t)b * 3 * N_;
    const float* py = px + N_;
    const float* pz = py + N_;

    // Stage all target coords + squared norms in LDS (48KB + 16KB).
    for (int i = tid; i < N_; i += QPB) {
        float x = px[i], y = py[i], z = pz[i];
        s_p[i] = x; s_p[N_ + i] = y; s_p[2 * N_ + i] = z;
        s_sq[i] = x * x + y * y + z * z;
    }
    __syncthreads();

    const int  w    = tid >> 5;
    const int  lane = tid & 31;
    const int  m    = lane & 15;
    const bool hi   = lane >= 16;            // K-half select (ISA 16x4 f32 A/B layout)
    const int  qbase = qchunk * QPB + w * QPW;

    // Branch-free lane selectors: lo lanes read {x,y}, hi lanes read {z, 0}.
    const int   koff  = hi ? 2 * N_ : 0;     // which coord plane feeds VGPR0
    const float ymask = hi ? 0.0f : 1.0f;    // zeroes the K=3 pad in VGPR1

    // A tiles: two 16x4 f32 matrices (queries), pre-scaled by -2, K=3 padded to 4.
    v2f a0, a1;
    {
        int q0 = qbase + m;
        int q1 = qbase + 16 + m;
        a0.x = -2.0f * s_p[koff + q0];
        a0.y = -2.0f * ymask * s_p[N_ + q0];
        a1.x = -2.0f * s_p[koff + q1];
        a1.y = -2.0f * ymask * s_p[N_ + q1];
    }

    // Register-resident sorted top-21 (ascending distance; stable on index via strict <).
    float best_d[KNN_];
    int   best_i[KNN_];
#pragma unroll
    for (int j = 0; j < KNN_; ++j) { best_d[j] = 3.4e38f; best_i[j] = N_; }

    const float sqQ = s_sq[qbase + lane];    // this lane's query norm (row = lane)

    for (int tb = 0; tb < N_; tb += 16) {
        // B tile: 4x16 f32 (targets), mirrored K-split layout; straight-line DS loads.
        const int t = tb + m;
        v2f bt;
        bt.x = s_p[koff + t];
        bt.y = ymask * s_p[N_ + t];
        const float sqT = s_sq[t];           // norm of target column m

        v8f c0 = {};
        v8f c1 = {};
        // D = (-2*Q) x T^T  -> two 16x16 dot-product tiles
        c0 = __builtin_amdgcn_wmma_f32_16x16x4_f32(false, a0, false, bt,
                                                   (short)0, c0, false, false);
        c1 = __builtin_amdgcn_wmma_f32_16x16x4_f32(false, a1, false, bt,
                                                   (short)0, c1, false, false);

        // Scatter tiles to this wave's LDS scratch, adding target norms.
        // f32 C/D layout: VGPR v -> row (lane<16 ? v : v+8), col (lane&15).
#pragma unroll
        for (int v = 0; v < 8; ++v) {
            int row = hi ? (v + 8) : v;
            s_dt[w][row][m]      = c0[v] + sqT;
            s_dt[w][row + 16][m] = c1[v] + sqT;
        }
        // Wave-private LDS handoff: DS ops are in-order per wave; wait + compiler fence.
        asm volatile("s_wait_dscnt 0" ::: "memory");

        // Each lane owns query row 'lane'; scan 16 candidates, update top-21.
        // Keep this loop ROLLED: one copy of the insertion network (I-cache friendly).
#pragma unroll 1
        for (int cc = 0; cc < 16; ++cc) {
            float d = s_dt[w][lane][cc] + sqQ;
            d = fmaxf(d, 1e-12f);            // matches reference clip; self -> 1e-12
            if (d < best_d[KNN_ - 1]) {
                float cd = d; int ci = tb + cc;
#pragma unroll
                for (int j = 0; j < KNN_; ++j) {
                    if (cd < best_d[j]) {
                        float td = best_d[j]; int ti = best_i[j];
                        best_d[j] = cd; best_i[j] = ci;
                        cd = td; ci = ti;
                    }
                }
            }
        }
    }

    // Drop the nearest entry (self) and emit 20 neighbor indices.
    int q = qbase + lane;
    int* op = knn_idx + ((size_t)b * N_ + q) * KOUT;
#pragma unroll
    for (int k = 0; k < KOUT; ++k) op[k] = best_i[k + 1];
}

// Phase 2: out[b][c][n][k]; c<64 -> center, c>=64 -> feats[b][c-64][idx]-center.
// Grid: x = (N*KOUT)/256 chunks (contiguous n,k), y = channel, z = batch.
__global__ __launch_bounds__(256) void knn_gather_kernel(
    const float* __restrict__ feats, const int* __restrict__ knn_idx,
    float* __restrict__ out)
{
    const int t = blockIdx.x * 256 + threadIdx.x;   // 0 .. N_*KOUT-1
    const int k = t % KOUT;
    const int n = t / KOUT;
    const int c = blockIdx.y;                       // 0 .. 2C-1
    const int b = blockIdx.z;
    const int cc = c & (C_ - 1);

    const float* frow = feats + ((size_t)b * C_ + cc) * N_;
    const float  center = frow[n];

    float val;
    if (c < C_) {
        val = center;
    } else {
        int j = knn_idx[((size_t)b * N_ + n) * KOUT + k];
        val = frow[j] - center;                     // gather: 16KB row, L2-resident
    }
    out[(((size_t)b * (2 * C_) + c) * N_) * KOUT + t] = val;  // fully coalesced
}

extern "C" void kernel_launch(void* const* d_in, const int* in_sizes, int n_in,
                              void* d_out, int out_size, void* d_ws, size_t ws_size,
                              hipStream_t stream)
{
    const float* points = (const float*)d_in[0];   // (B,3,N) f32
    const float* feats  = (const float*)d_in[1];   // (B,C,N,1) f32
    float* out = (float*)d_out;                    // (B,2C,N,KOUT) f32
    int* knn_idx = (int*)d_ws;                     // B*N*KOUT ints = 2.62 MB scratch

    dim3 g1(B_ * (N_ / QPB));                      // 256 blocks x 128 threads
    knn_topk_kernel<<<g1, dim3(QPB), 0, stream>>>(points, knn_idx);

    dim3 g2((N_ * KOUT) / 256, 2 * C_, B_);        // 320 x 128 x 8 blocks
    knn_gather_kernel<<<g2, dim3(256), 0, stream>>>(feats, knn_idx, out);
}